// MaceModel_72164040507913
// MI455X (gfx1250) — compile-verified
//
#include <hip/hip_runtime.h>
#include <hip/hip_bf16.h>
#include <cstdint>
#include <cstddef>

typedef __attribute__((ext_vector_type(2))) float v2f;
typedef __attribute__((ext_vector_type(8))) float v8f;

#define HS 64
#define HH 256
#define GG 16
#define ECHUNK 65536

#define KTILE 64
#define BN 64
#define KPAD (KTILE + 2)   // even -> all ds_load_b64 stay 8B aligned

#if defined(__has_builtin)
#if __has_builtin(__builtin_amdgcn_global_load_async_to_lds_b32) && \
    __has_builtin(__builtin_amdgcn_s_wait_asynccnt)
#define HAVE_ASYNC_LDS 1
#endif
#endif

typedef __attribute__((address_space(1))) int g_int;
typedef __attribute__((address_space(3))) int l_int;

static inline int cdiv(int a, int b) { return (a + b - 1) / b; }

// ---------------------------------------------------------------------------
// fp32 GEMM via V_WMMA_F32_16X16X4_F32 with LDS-staged B panel.
// C[M,N] = act(A[M,K] @ B[K,N] + bias)  (acc=1: C += A@B, no bias/act)
// Row-major. N % 64 == 0 (all uses: 64/128/192/256). K % 4 == 0. M arbitrary.
// Block = 8 waves; each wave owns a 16x64 C tile (4 v8f accumulators).
// B panel (KTILE x 64) staged col-major in LDS (async-to-LDS when available),
// so each B fragment is one ds_load_b64; A fragment is one global b64.
// ---------------------------------------------------------------------------
__global__ void gemm_wmma_f32(const float* __restrict__ A,
                              const float* __restrict__ B,
                              float* __restrict__ C,
                              int M, int N, int K,
                              const float* __restrict__ bias,
                              int act, int acc) {
  __shared__ float Bl[BN * KPAD];

  const int tid  = threadIdx.x;
  const int wave = tid >> 5;
  const int lane = tid & 31;
  const int mt   = blockIdx.x * (blockDim.x >> 5) + wave;
  const int row0 = mt << 4;
  const int n0   = blockIdx.y * 64;
  const int l    = lane & 15;      // M index (A) / N index (B,C)
  const int h    = lane >> 4;      // K-pair selector / row-half selector

  v8f accs[4] = {v8f{}, v8f{}, v8f{}, v8f{}};

  int arow = row0 + l;
  if (arow >= M) arow = M - 1;     // clamp: only feeds rows we never store
  const float* abase = A + (size_t)arow * K + 2 * h;

  for (int k0 = 0; k0 < K; k0 += KTILE) {
    const int kt = (K - k0 < KTILE) ? (K - k0) : KTILE;
    const int total = kt * BN;

    // ---- stage B[k0..k0+kt) x [n0..n0+64) into LDS, col-major ----
#if HAVE_ASYNC_LDS
    for (int idx = tid; idx < total; idx += 256) {
      const int n  = idx & (BN - 1);
      const int kk = idx >> 6;
      __builtin_amdgcn_global_load_async_to_lds_b32(
          (g_int*)(B + (size_t)(k0 + kk) * N + n0 + n),
          (l_int*)(&Bl[n * KPAD + kk]),
          0, 0);
    }
    __builtin_amdgcn_s_wait_asynccnt(0);
#else
    for (int idx = tid; idx < total; idx += 256) {
      const int n  = idx & (BN - 1);
      const int kk = idx >> 6;
      Bl[n * KPAD + kk] = B[(size_t)(k0 + kk) * N + n0 + n];
    }
#endif
    __syncthreads();

    // prefetch next B tile while we compute on this one
    if (k0 + KTILE < K)
      __builtin_prefetch(B + (size_t)(k0 + KTILE) * N + n0 + (tid & 63), 0, 1);

    // ---- compute ----
    const float* ap  = abase + k0;
    const float* blp = &Bl[l * KPAD + 2 * h];
    for (int kk = 0; kk < kt; kk += 4) {
      v2f a = *(const v2f*)(ap + kk);
#pragma unroll
      for (int t = 0; t < 4; ++t) {
        v2f b = *(const v2f*)(blp + kk + t * (16 * KPAD));
        accs[t] = __builtin_amdgcn_wmma_f32_16x16x4_f32(
            false, a, false, b, (short)0, accs[t], false, false);
      }
    }
    __syncthreads();
  }

  // D/C layout: lane l = column, VGPR j = row j + 8h (ISA 7.12.2)
#pragma unroll
  for (int t = 0; t < 4; ++t) {
    const int col = n0 + t * 16 + l;
    const float bv = bias ? bias[col] : 0.0f;
#pragma unroll
    for (int j = 0; j < 8; ++j) {
      const int row = row0 + j + 8 * h;
      if (row >= M) continue;
      float v = accs[t][j] + bv;
      if (act == 1)      v = v / (1.0f + __expf(-v));    // silu
      else if (act == 2) v = 1.0f / (1.0f + __expf(-v)); // sigmoid
      float* cp = C + (size_t)row * N + col;
      if (acc) v += *cp;
      *cp = v;
    }
  }
}

// ---------------------------------------------------------------------------
// Elementwise / gather / scatter kernels
// ---------------------------------------------------------------------------
__global__ void zero_kernel(float* p, size_t n) {
  size_t i = (size_t)blockIdx.x * blockDim.x + threadIdx.x;
  size_t stride = (size_t)gridDim.x * blockDim.x;
  for (; i < n; i += stride) p[i] = 0.0f;
}

__global__ void edge_u_kernel(const long long* __restrict__ ei,
                              const float* __restrict__ pos,
                              float* __restrict__ u, int E) {
  int e = blockIdx.x * blockDim.x + threadIdx.x;
  if (e >= E) return;
  long long s = ei[e], d = ei[(size_t)E + e];
  float vx = pos[s * 3 + 0] - pos[d * 3 + 0];
  float vy = pos[s * 3 + 1] - pos[d * 3 + 1];
  float vz = pos[s * 3 + 2] - pos[d * 3 + 2];
  float nrm = sqrtf(vx * vx + vy * vy + vz * vz) + 1e-9f;
  u[e * 3 + 0] = vx / nrm;
  u[e * 3 + 1] = vy / nrm;
  u[e * 3 + 2] = vz / nrm;
}

// layer0: v = pos[:,None,:], Wv_in (1,64) -> vin[i][n][d] = pos[n][i]*Wv[d]
__global__ void vin_from_pos_kernel(const float* __restrict__ pos,
                                    const float* __restrict__ Wv,
                                    float* __restrict__ vin,
                                    int N, size_t plane) {
  int idx = blockIdx.x * blockDim.x + threadIdx.x;
  if (idx >= N * HS) return;
  int n = idx >> 6, d = idx & 63;
  float w = Wv[d];
  vin[idx]             = pos[n * 3 + 0] * w;
  vin[plane + idx]     = pos[n * 3 + 1] * w;
  vin[2 * plane + idx] = pos[n * 3 + 2] * w;
}

__global__ void ew_mul_kernel(const float* __restrict__ a,
                              const float* __restrict__ b,
                              float* __restrict__ o, size_t n) {
  size_t i = (size_t)blockIdx.x * blockDim.x + threadIdx.x;
  if (i < n) o[i] = a[i] * b[i];
}

__global__ void ew_pow23_kernel(const float* __restrict__ s,
                                float* __restrict__ s2,
                                float* __restrict__ s3, size_t n) {
  size_t i = (size_t)blockIdx.x * blockDim.x + threadIdx.x;
  if (i < n) { float v = s[i]; s2[i] = v * v; s3[i] = v * v * v; }
}

// Per edge, per channel: compute messages and scatter-add into S / V planes.
__global__ void message_kernel(const long long* __restrict__ ei, int E,
                               int e0, int elen,
                               const float* __restrict__ R,      // [elen,192]
                               const float* __restrict__ s_in,   // [N,64]
                               const float* __restrict__ vin,    // 3 planes
                               size_t plane,
                               const float* __restrict__ u,      // [E,3]
                               float* __restrict__ S,            // [N,64]
                               float* __restrict__ V) {          // 3 planes
  int t = blockIdx.x * blockDim.x + threadIdx.x;
  int er = t >> 6;
  if (er >= elen) return;
  int c = t & 63;
  int e = e0 + er;
  long long s = ei[e], d = ei[(size_t)E + e];
  size_t rb = (size_t)er * 192;
  float r0 = R[rb + c], r1 = R[rb + 64 + c], r2 = R[rb + 128 + c];
  size_t so = (size_t)s * HS + c;
  float ss = s_in[so];
  float v0 = vin[so], v1 = vin[plane + so], v2 = vin[2 * plane + so];
  float ux = u[e * 3 + 0], uy = u[e * 3 + 1], uz = u[e * 3 + 2];
  float dot = v0 * ux + v1 * uy + v2 * uz;
  float ms   = (r0 * ss + r1 * dot) * 0.1f;   // / AVG_N
  float base = r2 * ss * 0.1f;
  size_t dofs = (size_t)d * HS + c;
  atomicAdd(&S[dofs], ms);
  atomicAdd(&V[dofs], base * ux);
  atomicAdd(&V[plane + dofs], base * uy);
  atomicAdd(&V[2 * plane + dofs], base * uz);
}

// node_feats = concat(s, v.reshape(N,3*HS)); v (n,c,i) -> col 64 + c*3 + i
__global__ void concat_kernel(const float* __restrict__ s,
                              const float* __restrict__ v, size_t plane,
                              float* __restrict__ nf, int N) {
  int idx = blockIdx.x * blockDim.x + threadIdx.x;
  if (idx >= N * HS) return;
  int n = idx >> 6, c = idx & 63;
  size_t nb = (size_t)n * HH;
  nf[nb + c] = s[idx];
  nf[nb + 64 + c * 3 + 0] = v[idx];
  nf[nb + 64 + c * 3 + 1] = v[plane + idx];
  nf[nb + 64 + c * 3 + 2] = v[2 * plane + idx];
}

// w[n] = sigmoid(nf[n] . Wpool + bpool)   (one wave32 per node)
__global__ void poolw_kernel(const float* __restrict__ nf,
                             const float* __restrict__ Wpool,
                             const float* __restrict__ bpool,
                             float* __restrict__ w, int N) {
  int gi = blockIdx.x * blockDim.x + threadIdx.x;
  int node = gi >> 5, lane = threadIdx.x & 31;
  if (node >= N) return;
  float sum = 0.0f;
  const float* row = nf + (size_t)node * HH;
  for (int k = lane; k < HH; k += 32) sum += row[k] * Wpool[k];
  for (int off = 16; off; off >>= 1) sum += __shfl_xor(sum, off, 32);
  if (lane == 0) w[node] = 1.0f / (1.0f + __expf(-(sum + bpool[0])));
}

// g_h[b] += w[n] * nf[n]; batch sorted -> per-column running sum, atomic
// flush only on group change. One block = 256 cols x 256-node slab.
__global__ void pool_segsum_kernel(const float* __restrict__ nf,
                                   const float* __restrict__ w,
                                   const long long* __restrict__ batch,
                                   float* __restrict__ g, int N) {
  int col = threadIdx.x;
  int n0 = blockIdx.x * 256;
  if (n0 >= N) return;
  int n1 = n0 + 256; if (n1 > N) n1 = N;
  long long cur = batch[n0];
  float acc = 0.0f;
  for (int n = n0; n < n1; ++n) {
    long long b = batch[n];
    if (b != cur) {
      atomicAdd(&g[(size_t)cur * HH + col], acc);
      acc = 0.0f; cur = b;
    }
    acc += w[n] * nf[(size_t)n * HH + col];
  }
  atomicAdd(&g[(size_t)cur * HH + col], acc);
}

// node_logits[n] = hid[n] @ Wn2 + bn2   (hid: N x 128, Wn2: 128 x 3)
__global__ void node_logits_kernel(const float* __restrict__ hid,
                                   const float* __restrict__ Wn2,
                                   const float* __restrict__ bn2,
                                   float* __restrict__ out, int N) {
  int gi = blockIdx.x * blockDim.x + threadIdx.x;
  int node = gi >> 5, lane = threadIdx.x & 31;
  if (node >= N) return;
  float s0 = 0.0f, s1 = 0.0f, s2 = 0.0f;
  const float* row = hid + (size_t)node * 128;
  for (int k = lane; k < 128; k += 32) {
    float hv = row[k];
    s0 += hv * Wn2[k * 3 + 0];
    s1 += hv * Wn2[k * 3 + 1];
    s2 += hv * Wn2[k * 3 + 2];
  }
  for (int off = 16; off; off >>= 1) {
    s0 += __shfl_xor(s0, off, 32);
    s1 += __shfl_xor(s1, off, 32);
    s2 += __shfl_xor(s2, off, 32);
  }
  if (lane == 0) {
    out[node * 3 + 0] = s0 + bn2[0];
    out[node * 3 + 1] = s1 + bn2[1];
    out[node * 3 + 2] = s2 + bn2[2];
  }
}

// logits = silu(g_h @ Wp1 + bp1) @ Wp2 + bp2    (tiny: single block)
__global__ void graph_mlp_kernel(const float* __restrict__ g,
                                 const float* __restrict__ Wp1,
                                 const float* __restrict__ bp1,
                                 const float* __restrict__ Wp2,
                                 const float* __restrict__ bp2,
                                 float* __restrict__ logits) {
  __shared__ float hid[GG * 128];
  for (int idx = threadIdx.x; idx < GG * 128; idx += blockDim.x) {
    int gi = idx >> 7, j = idx & 127;
    float sum = bp1[j];
    const float* grow = g + (size_t)gi * HH;
    for (int k = 0; k < HH; ++k) sum += grow[k] * Wp1[k * 128 + j];
    hid[idx] = sum / (1.0f + __expf(-sum));
  }
  __syncthreads();
  int wave = threadIdx.x >> 5, lane = threadIdx.x & 31;
  for (int gi = wave; gi < GG; gi += (int)(blockDim.x >> 5)) {
    float sum = 0.0f;
    for (int j = lane; j < 128; j += 32) sum += hid[gi * 128 + j] * Wp2[j];
    for (int off = 16; off; off >>= 1) sum += __shfl_xor(sum, off, 32);
    if (lane == 0) logits[gi] = sum + bp2[0];
  }
}

// ---------------------------------------------------------------------------
// Host orchestration
// ---------------------------------------------------------------------------
static void gemm(hipStream_t st, const float* A, const float* B, float* C,
                 int M, int N, int K, const float* bias, int act, int acc) {
  dim3 grid(cdiv(M, 128), N / 64);
  gemm_wmma_f32<<<grid, 256, 0, st>>>(A, B, C, M, N, K, bias, act, acc);
}

static void zero_f(hipStream_t st, float* p, size_t n) {
  int blocks = (int)((n + 255) / 256); if (blocks > 4096) blocks = 4096;
  zero_kernel<<<blocks, 256, 0, st>>>(p, n);
}

extern "C" void kernel_launch(void* const* d_in, const int* in_sizes, int n_in,
                              void* d_out, int out_size, void* d_ws, size_t ws_size,
                              hipStream_t stream) {
  (void)n_in; (void)out_size; (void)ws_size;
  const int N = in_sizes[2] / 3;   // pos
  const int E = in_sizes[1] / 2;   // edge_index

  const float*     x          = (const float*)d_in[0];
  const long long* edge_index = (const long long*)d_in[1];
  const float*     pos        = (const float*)d_in[2];
  const float*     edge_attr  = (const float*)d_in[3];
  const long long* batch      = (const long long*)d_in[4];
  const float* W_in   = (const float*)d_in[5];
  const float* b_in   = (const float*)d_in[6];
  const float* W_attr = (const float*)d_in[7];
  const float* b_attr = (const float*)d_in[8];
  // per-layer params (insertion order): Ws_in,Wv_in,Wr1,br1,Wr2,Wa,W1,W2,W3,Wv1,Wsc_s,Wsc_v
  const int LP[2] = {9, 21};
  const float* Wpool = (const float*)d_in[33];
  const float* bpool = (const float*)d_in[34];
  const float* Wn1   = (const float*)d_in[35];
  const float* bn1   = (const float*)d_in[36];
  const float* Wn2   = (const float*)d_in[37];
  const float* bn2   = (const float*)d_in[38];
  const float* Wp1   = (const float*)d_in[39];
  const float* bp1   = (const float*)d_in[40];
  const float* Wp2   = (const float*)d_in[41];
  const float* bp2   = (const float*)d_in[42];

  // ---- workspace layout (floats) ----
  const size_t plane = (size_t)N * HS;
  float* ws = (float*)d_ws;
  size_t o = 0;
  float* u_buf   = ws + o; o += (size_t)E * 3;
  float* attrs   = ws + o; o += (size_t)N * HH;
  float* h_buf   = ws + o; o += (size_t)N * HH;
  float* a_buf   = ws + o; o += plane;
  float* sin_buf = ws + o; o += plane;
  float* vin_buf = ws + o; o += 3 * plane;
  float* S_buf   = ws + o; o += plane;
  float* V_buf   = ws + o; o += 3 * plane;
  float* s_cur   = ws + o; o += plane;
  float* v_cur   = ws + o; o += 3 * plane;
  size_t scr_sz = (size_t)ECHUNK * (64 + 192);
  if (4 * plane > scr_sz) scr_sz = 4 * plane;
  if ((size_t)N * 128 > scr_sz) scr_sz = (size_t)N * 128;
  float* scr     = ws + o; o += scr_sz;
  float* w_buf   = ws + o; o += (size_t)N;

  float* t_chunk = scr;
  float* R_chunk = scr + (size_t)ECHUNK * 64;
  float* S2 = scr;            // node-update phase (chunk bufs dead by then)
  float* S3 = scr + plane;
  float* P4 = scr + 2 * plane;
  float* TMP = scr + 3 * plane;
  float* hid = scr;           // head phase

  // ---- output layout ----
  float* out_f       = (float*)d_out;
  float* logits      = out_f;
  float* node_logits = out_f + GG;
  float* g_h         = out_f + GG + (size_t)N * 3;
  float* nf          = g_h + (size_t)GG * HH;

  // ---- geometry + input embeddings ----
  edge_u_kernel<<<cdiv(E, 256), 256, 0, stream>>>(edge_index, pos, u_buf, E);
  gemm(stream, x, W_in,   h_buf, N, HH, 128, b_in,   0, 0);  // h
  gemm(stream, x, W_attr, attrs, N, HH, 128, b_attr, 0, 0);  // node_attrs

  // ---- 2 MACE layers ----
  for (int l = 0; l < 2; ++l) {
    const int p = LP[l];
    const float* Ws_in  = (const float*)d_in[p + 0];
    const float* Wv_in  = (const float*)d_in[p + 1];
    const float* Wr1    = (const float*)d_in[p + 2];
    const float* br1    = (const float*)d_in[p + 3];
    const float* Wr2    = (const float*)d_in[p + 4];
    const float* Wa     = (const float*)d_in[p + 5];
    const float* W1     = (const float*)d_in[p + 6];
    const float* W2     = (const float*)d_in[p + 7];
    const float* W3     = (const float*)d_in[p + 8];
    const float* Wv1    = (const float*)d_in[p + 9];
    const float* Wsc_s  = (const float*)d_in[p + 10];
    const float* Wsc_v  = (const float*)d_in[p + 11];
    const int c0 = (l == 0) ? HH : HS;
    const float* s_src_mat = (l == 0) ? h_buf : s_cur;

    // s_in = s @ Ws_in
    gemm(stream, s_src_mat, Ws_in, sin_buf, N, HS, c0, nullptr, 0, 0);
    // v_in
    if (l == 0) {
      vin_from_pos_kernel<<<cdiv(N * HS, 256), 256, 0, stream>>>(
          pos, Wv_in, vin_buf, N, plane);
    } else {
      for (int i = 0; i < 3; ++i)
        gemm(stream, v_cur + i * plane, Wv_in, vin_buf + i * plane,
             N, HS, HS, nullptr, 0, 0);
    }
    // a = node_attrs @ Wa
    gemm(stream, attrs, Wa, a_buf, N, HS, HH, nullptr, 0, 0);

    // zero accumulators
    zero_f(stream, S_buf, plane);
    zero_f(stream, V_buf, 3 * plane);

    // edge MLP + message scatter, chunked
    for (int e0 = 0; e0 < E; e0 += ECHUNK) {
      int len = E - e0; if (len > ECHUNK) len = ECHUNK;
      gemm(stream, edge_attr + (size_t)e0 * 32, Wr1, t_chunk,
           len, HS, 32, br1, 1, 0);                       // silu(..+br1)
      gemm(stream, t_chunk, Wr2, R_chunk, len, 192, HS, nullptr, 0, 0);
      message_kernel<<<cdiv(len * HS, 256), 256, 0, stream>>>(
          edge_index, E, e0, len, R_chunk, sin_buf, vin_buf, plane,
          u_buf, S_buf, V_buf);
    }

    // node update
    ew_pow23_kernel<<<cdiv((int)plane, 256), 256, 0, stream>>>(S_buf, S2, S3, plane);
    ew_mul_kernel<<<cdiv((int)plane, 256), 256, 0, stream>>>(a_buf, sin_buf, P4, plane);
    gemm(stream, S_buf, W1,    s_cur, N, HS, HS, nullptr, 0, 0);
    gemm(stream, S2,    W2,    s_cur, N, HS, HS, nullptr, 0, 1);
    gemm(stream, S3,    W3,    s_cur, N, HS, HS, nullptr, 0, 1);
    gemm(stream, P4,    Wsc_s, s_cur, N, HS, HS, nullptr, 0, 1);
    for (int i = 0; i < 3; ++i) {
      ew_mul_kernel<<<cdiv((int)plane, 256), 256, 0, stream>>>(
          V_buf + i * plane, a_buf, TMP, plane);
      gemm(stream, TMP,                 Wv1,   v_cur + i * plane, N, HS, HS, nullptr, 0, 0);
      gemm(stream, vin_buf + i * plane, Wsc_v, v_cur + i * plane, N, HS, HS, nullptr, 0, 1);
    }
  }

  // ---- heads ----
  concat_kernel<<<cdiv(N * HS, 256), 256, 0, stream>>>(s_cur, v_cur, plane, nf, N);

  zero_f(stream, g_h, (size_t)GG * HH);
  poolw_kernel<<<cdiv(N * 32, 256), 256, 0, stream>>>(nf, Wpool, bpool, w_buf, N);
  pool_segsum_kernel<<<cdiv(N, 256), 256, 0, stream>>>(nf, w_buf, batch, g_h, N);

  gemm(stream, nf, Wn1, hid, N, 128, HH, bn1, 1, 0);        // silu(nf@Wn1+bn1)
  node_logits_kernel<<<cdiv(N * 32, 256), 256, 0, stream>>>(hid, Wn2, bn2,
                                                            node_logits, N);
  graph_mlp_kernel<<<1, 256, 0, stream>>>(g_h, Wp1, bp1, Wp2, bp2, logits);
}